// RoIAlignAdaMax_73349451481354
// MI455X (gfx1250) — compile-verified
//
#include <hip/hip_runtime.h>
#include <hip/hip_bf16.h>
#include <stdint.h>

// Problem constants (fixed by the reference harness)
#define BQ   4
#define CQ   512
#define HQ   96
#define WQ   96
#define AHF  11          // fine grid 11x11
#define NFINE (AHF*AHF)  // 121

// Tiling
#define CH        8            // channels per block
#define RW        40           // tile width  (covers max roi footprint: <=37)
#define RH        40           // tile height
#define TILE_SLAB (RH*RW)      // 1600 floats per channel
#define NTHREADS  128          // 4 waves (wave32)

typedef unsigned int u32x4 __attribute__((ext_vector_type(4)));
typedef int          i32x8 __attribute__((ext_vector_type(8)));
typedef int          i32x4 __attribute__((ext_vector_type(4)));

#if __has_builtin(__builtin_amdgcn_tensor_load_to_lds) && __has_builtin(__builtin_amdgcn_s_wait_tensorcnt)
#define HAVE_TDM 1
#else
#define HAVE_TDM 0
#endif

__global__ __launch_bounds__(NTHREADS)
void RoIAlignAdaMax_73349451481354_kernel(const float* __restrict__ feat,
                                          const float* __restrict__ rois,
                                          float* __restrict__ out)
{
    extern __shared__ float smem[];
    float* tile = smem;                    // [CH][RH][RW] packed
    float* fine = smem + CH * TILE_SLAB;   // [121]

    const int tid   = threadIdx.x;
    const int chunk = blockIdx.x;          // 0 .. C/CH-1
    const int r     = blockIdx.y;          // roi index
    const int c0    = chunk * CH;

    // ---- roi geometry (uniform per block; scalar loads) --------------------
    const float* rp = rois + (size_t)r * 5;
    int   bi = (int)rp[0];
    float x1 = rp[1] * 0.125f, y1 = rp[2] * 0.125f;
    float x2 = rp[3] * 0.125f, y2 = rp[4] * 0.125f;
    float roi_w = fmaxf(x2 - x1, 0.0f), roi_h = fmaxf(y2 - y1, 0.0f);
    float bin_w = roi_w / 10.0f,        bin_h = roi_h / 10.0f;
    float str_w = fmaxf(1.0f, rintf(bin_w / 3.0f));   // round-half-even, matches jnp.round
    float str_h = fmaxf(1.0f, rintf(bin_h / 3.0f));

    int x0 = (int)floorf(x1 - str_w) - 1;
    x0 = x0 < 0 ? 0 : (x0 > WQ - RW ? WQ - RW : x0);
    int y0 = (int)floorf(y1 - str_h) - 1;
    y0 = y0 < 0 ? 0 : (y0 > HQ - RH ? HQ - RH : y0);

    bi = __builtin_amdgcn_readfirstlane(bi);
    x0 = __builtin_amdgcn_readfirstlane(x0);
    y0 = __builtin_amdgcn_readfirstlane(y0);

#if HAVE_TDM
    // ---- per-wave Tensor-Data-Mover load: 40x40 x (CH/4) channels ----------
    {
        const int wv   = tid >> 5;           // wave id 0..3 (wave-uniform)
        const int WCH  = CH / 4;             // channels per wave
        const int c0w  = c0 + wv * WCH;

        unsigned long long ga = (unsigned long long)(size_t)feat
            + 4ull * ((((unsigned long long)bi * CQ + (unsigned)c0w) * HQ
                       + (unsigned)y0) * WQ + (unsigned)x0);
        unsigned int lds_base = (unsigned int)(size_t)(void*)tile
                              + (unsigned int)(wv * WCH * TILE_SLAB * 4);
        unsigned int td0 = (unsigned)(WQ - x0);      // remaining extent dim0
        unsigned int td1 = (unsigned)(HQ - y0);      // remaining extent dim1
        unsigned int td2 = (unsigned)(CQ - c0w);     // remaining extent dim2
        unsigned int t0s = (unsigned)WQ;             // line stride (dim0 -> next row)
        unsigned int t1s = (unsigned)(HQ * WQ);      // channel stride (dim1 -> next ch)

        u32x4 g0;
        g0[0] = 1u;                                          // count=1, user descriptor
        g0[1] = lds_base;                                    // lds_addr (bytes)
        g0[2] = (unsigned)(ga & 0xFFFFFFFFull);              // global_addr lo
        g0[3] = (unsigned)((ga >> 32) & 0x1FFFFFFull) | (2u << 30);  // hi bits | type=2

        i32x8 g1;
        g1[0] = (int)(2u << 16);                             // data_size = 4B
        g1[1] = (int)((td0 & 0xFFFFu) << 16);                // tensor_dim0 [79:48]
        g1[2] = (int)(((td0 >> 16) & 0xFFFFu) | ((td1 & 0xFFFFu) << 16));
        g1[3] = (int)(((td1 >> 16) & 0xFFFFu) | ((unsigned)RW << 16));  // tile_dim0
        g1[4] = (int)((unsigned)RH | ((unsigned)WCH << 16)); // tile_dim1 | tile_dim2
        g1[5] = (int)t0s;                                    // dim0_stride lo32 (hi16 = 0)
        g1[6] = (int)((t1s & 0xFFFFu) << 16);                // dim0_stride hi16=0 | dim1_stride lo16
        g1[7] = (int)(t1s >> 16);                            // dim1_stride hi32

        i32x4 g2; g2[0] = (int)td2; g2[1] = 0; g2[2] = 0; g2[3] = 0;  // tensor_dim2; tile_dim3=0
        i32x4 g3; g3[0] = 0; g3[1] = 0; g3[2] = 0; g3[3] = 0;

        i32x8 gz;                                            // extra arg (clang-23 6-arg form)
        gz[0] = 0; gz[1] = 0; gz[2] = 0; gz[3] = 0;
        gz[4] = 0; gz[5] = 0; gz[6] = 0; gz[7] = 0;

        __builtin_amdgcn_tensor_load_to_lds(g0, g1, g2, g3, gz, 0);
    }
#endif

    // ---- per-lane sample metadata (channel-invariant, kept in VGPRs) -------
    const int  fb     = tid;
    const bool active = fb < NFINE;
    int   ulo[9];
    float w00[9], w01[9], w10[9], w11[9];
    float cntm = 1.0f;
    bool  cok  = false;

    if (active) {
        int   fh = fb / AHF, fw = fb % AHF;
        float hc = y1 + (float)fh * bin_h;
        float wc = x1 + (float)fw * bin_w;
        cok = (hc >= 0.0f) && (hc < (float)HQ) && (wc >= 0.0f) && (wc < (float)WQ);
        float cnt = 0.0f;
#pragma unroll
        for (int s = 0; s < 9; ++s) {
            float sh = (float)(s / 3 - 1), sw = (float)(s % 3 - 1);
            float ch = hc + sh * str_h;
            float cw = wc + sw * str_w;
            bool  valid = (ch >= 0.0f) && (ch < (float)HQ) &&
                          (cw >= 0.0f) && (cw < (float)WQ);
            float hs = fminf(fmaxf(floorf(ch), 0.0f), (float)(HQ - 2));
            float ws = fminf(fmaxf(floorf(cw), 0.0f), (float)(WQ - 2));
            float hr = ch - hs, wr = cw - ws;
            float m  = valid ? 1.0f : 0.0f;
            cnt += m;
            float oh = 1.0f - hr, ow = 1.0f - wr;
            w00[s] = m * oh * ow;
            w01[s] = m * oh * wr;
            w10[s] = m * hr * ow;
            w11[s] = m * hr * wr;
            ulo[s] = ((int)hs - y0) * RW + ((int)ws - x0);
        }
        cntm = fmaxf(cnt, 1.0f);
    }

#if HAVE_TDM
    __builtin_amdgcn_s_wait_tensorcnt(0);
#else
    // Fallback: cooperative coalesced load of the tile
    for (int i = tid; i < CH * TILE_SLAB; i += NTHREADS) {
        int c = i / TILE_SLAB, p = i % TILE_SLAB;
        int yy = p / RW, xx = p % RW;
        tile[i] = feat[(((size_t)bi * CQ + (c0 + c)) * HQ + (y0 + yy)) * WQ + (x0 + xx)];
    }
#endif
    __syncthreads();

    // ---- main loop: per channel, 121 lanes build fine grid, 25 lanes pool --
#pragma unroll 1
    for (int c = 0; c < CH; ++c) {
        if (active) {
            const float* tc = tile + c * TILE_SLAB;
            float acc = 0.0f;
#pragma unroll
            for (int s = 0; s < 9; ++s) {
                int u = ulo[s];
                float f00 = tc[u],      f01 = tc[u + 1];          // -> ds_read2_b32
                float f10 = tc[u + RW], f11 = tc[u + RW + 1];     // -> ds_read2_b32
                acc = fmaf(w00[s], f00,
                      fmaf(w01[s], f01,
                      fmaf(w10[s], f10,
                      fmaf(w11[s], f11, acc))));
            }
            fine[fb] = cok ? (acc / cntm) : 0.0f;
        }
        __syncthreads();

        if (tid < 25) {
            int ph = tid / 5, pw = tid % 5;
            const float* fr = fine + (2 * ph) * AHF + 2 * pw;
            float m = fr[0];
            m = fmaxf(m, fr[1]);           m = fmaxf(m, fr[2]);
            m = fmaxf(m, fr[AHF + 0]);     m = fmaxf(m, fr[AHF + 1]);     m = fmaxf(m, fr[AHF + 2]);
            m = fmaxf(m, fr[2 * AHF + 0]); m = fmaxf(m, fr[2 * AHF + 1]); m = fmaxf(m, fr[2 * AHF + 2]);
            out[((size_t)r * CQ + (c0 + c)) * 25 + tid] = m;   // 25 contiguous floats
        }
        __syncthreads();
    }
}

extern "C" void kernel_launch(void* const* d_in, const int* in_sizes, int n_in,
                              void* d_out, int out_size, void* d_ws, size_t ws_size,
                              hipStream_t stream)
{
    const float* feat = (const float*)d_in[0];
    const float* rois = (const float*)d_in[1];
    float*       out  = (float*)d_out;
    const int R = in_sizes[1] / 5;

    dim3 grid(CQ / CH, R);
    dim3 block(NTHREADS);
    size_t shmem = (size_t)(CH * TILE_SLAB + 128) * sizeof(float); // tile + fine buffer
    hipLaunchKernelGGL(RoIAlignAdaMax_73349451481354_kernel,
                       grid, block, shmem, stream, feat, rois, out);
}